// FlaxAttentionModule_33775622815897
// MI455X (gfx1250) — compile-verified
//
#include <hip/hip_runtime.h>
#include <stdint.h>

// ---------------------------------------------------------------------------
// KV-cache scatter + int8 (qk=32) re-quantization + mask build for MI455X.
// Memory-bound: ~1.05 GB traffic -> ~45us floor @ 23.3 TB/s. This revision
// trims VALU (co-limiting) via compile-time address decomposition and DPP8
// lane reductions instead of DS shuffles.
// CDNA5 paths: global_load_async_to_lds_b128 (ASYNCcnt, NT hint) double
// buffer, DPP8 wave32 reduction, NT b128 stores.
// Fixed-shape launch: 8192 blocks x 256 thr = 65536 waves x 16 rows each.
// ---------------------------------------------------------------------------

typedef float v4f __attribute__((ext_vector_type(4)));

#define B_  2
#define S_  8192
#define H_  32
#define D_  128
#define Q_  16
#define BSH (B_ * S_ * H_)                // 524288 rows per tensor
#define TOTAL_ROWS (2 * BSH)              // 1048576: K rows then V rows
#define NBLOCKS 8192
#define WAVES_PER_BLK 8
#define STRIDE (NBLOCKS * WAVES_PER_BLK)  // 65536 waves
#define ITER (TOTAL_ROWS / STRIDE)        // exactly 16 rows per wave
#define ROWSTEP ((size_t)STRIDE * D_)     // float offset between a wave's rows

// ---- CDNA5 async copy: global -> LDS, 16B per lane, non-temporal ----------
__device__ __forceinline__ void async_load_b128(uint32_t lds_byte_addr,
                                                const void* gptr) {
  asm volatile("global_load_async_to_lds_b128 %0, %1, off th:TH_LOAD_NT"
               :
               : "v"(lds_byte_addr), "v"((uint64_t)(uintptr_t)gptr)
               : "memory");
}

template <int N>
__device__ __forceinline__ void wait_asynccnt() {
#if __has_builtin(__builtin_amdgcn_s_wait_asynccnt)
  __builtin_amdgcn_s_wait_asynccnt(N);
#else
  asm volatile("s_wait_asynccnt %0" ::"i"(N) : "memory");
#endif
}

// ---- DPP8 xor-style max reduction within groups of 8 lanes ----------------
#define DPP8SEL(a, b, c, d, e, f, g, h)                                      \
  ((a) | ((b) << 3) | ((c) << 6) | ((d) << 9) | ((e) << 12) | ((f) << 15) | \
   ((g) << 18) | ((h) << 21))

template <int SEL>
__device__ __forceinline__ float max_dpp8(float m) {
  int p = __builtin_amdgcn_mov_dpp8(__builtin_bit_cast(int, m), SEL);
  return fmaxf(m, __builtin_bit_cast(float, p));
}

// absmax over the 32-elem group (8 lanes x float4) -> quantize -> dequantize.
__device__ __forceinline__ void process_row(v4f x, float* dst) {
  float m = fmaxf(fmaxf(fabsf(x.x), fabsf(x.y)),
                  fmaxf(fabsf(x.z), fabsf(x.w)));
  m = max_dpp8<DPP8SEL(1, 0, 3, 2, 5, 4, 7, 6)>(m);  // xor 1
  m = max_dpp8<DPP8SEL(2, 3, 0, 1, 6, 7, 4, 5)>(m);  // xor 2
  m = max_dpp8<DPP8SEL(4, 5, 6, 7, 0, 1, 2, 3)>(m);  // xor 4

  float scale = m * (1.0f / 127.0f);
  if (scale == 0.0f) scale = 1.0f;
  const float inv = 1.0f / scale;  // one precise div per group

  v4f q;
  q.x = fminf(fmaxf(rintf(x.x * inv), -127.0f), 127.0f);  // -> v_med3
  q.y = fminf(fmaxf(rintf(x.y * inv), -127.0f), 127.0f);
  q.z = fminf(fmaxf(rintf(x.z * inv), -127.0f), 127.0f);
  q.w = fminf(fmaxf(rintf(x.w * inv), -127.0f), 127.0f);
  v4f r4 = q * scale;

  __builtin_nontemporal_store(r4, (v4f*)dst);
}

// ---------------------------------------------------------------------------
// One wave = one row of 128 floats per pipeline slot; lane l owns [4l, 4l+4).
// Row_i = wave0 + i*65536, wave0 < 65536, so for unrolled literal i:
//   isV = i>=8, j = i&7, b = (i>>2)&1, s = s0 + (i&3)*2048,  s0 = wave0>>5,
//   h = wave0&31 -- all per-slot constants or loop invariants.
// Requires exactly <<<NBLOCKS, 256>>>.
// ---------------------------------------------------------------------------
__global__ __launch_bounds__(256) void kv_requant_kernel(
    const float* __restrict__ cached_key, const float* __restrict__ cached_value,
    const float* __restrict__ key, const float* __restrict__ value,
    const int* __restrict__ cur_index_p, float* __restrict__ out) {
  __shared__ v4f buf0[256];
  __shared__ v4f buf1[256];

  const int tid = threadIdx.x;
  const int lane = tid & 31;
  const int wave0 = (int)blockIdx.x * WAVES_PER_BLK + (tid >> 5);
  const int cur = cur_index_p[0];
  const int laneOff = lane * 4;

  // LDS byte addresses (flat LDS aperture truncates to addr[31:0]).
  const uint32_t lds0 = (uint32_t)(uintptr_t)&buf0[tid];
  const uint32_t lds1 = (uint32_t)(uintptr_t)&buf1[tid];

  // Loop-invariant decomposition of the row index.
  const int s0 = wave0 >> 5;        // seq position base (within 2048 chunk)
  const int h = wave0 & (H_ - 1);   // head
  const int u0 = s0 - cur;          // window test: (unsigned)(u0+k*2048) < 16

  // Loop-invariant base pointers, lane offset folded in.
  const float* ckB = cached_key + (size_t)wave0 * D_ + laneOff;
  const float* cvB = cached_value + (size_t)wave0 * D_ + laneOff;
  const float* kB = key + (size_t)h * D_ + laneOff;    // (B,Q,H,D), b,q to add
  const float* vB = value + (size_t)h * D_ + laneOff;
  float* outB = out + (size_t)wave0 * D_ + laneOff;

  // Per-slot source address; i must be a compile-time constant (unrolled).
  auto srcAddr = [&](int i) -> const float* {
    const int j = i & 7;
    const int k = i & 3;
    const int b = (i >> 2) & 1;
    const bool isV = i >= 8;
    const int u = u0 + k * 2048;  // s - cur
    const float* cachep = (isV ? cvB : ckB) + (size_t)j * ROWSTEP;
    const float* newp = (isV ? vB : kB) + (size_t)b * (Q_ * H_ * D_) +
                        ((long long)u << 12);  // u * H_*D_ (garbage if !inWin)
    return ((unsigned)u < (unsigned)Q_) ? newp : cachep;  // branchless select
  };

  // Prologue: first row in flight.
  async_load_b128(lds0, srcAddr(0));

#pragma unroll
  for (int i = 0; i < ITER; i += 2) {
    async_load_b128(lds1, srcAddr(i + 1));
    wait_asynccnt<1>();  // in-order completion: slot i has landed
    process_row(buf0[tid], outB + (size_t)i * ROWSTEP);

    if (i + 2 < ITER) {
      async_load_b128(lds0, srcAddr(i + 2));  // buf0 consumed above
      wait_asynccnt<1>();
    } else {
      wait_asynccnt<0>();
    }
    process_row(buf1[tid], outB + (size_t)(i + 1) * ROWSTEP);
  }
}

// ---------------------------------------------------------------------------
// attn_mask[b,0,q,s] = (s < cur + Q) && attention_mask[b,0,q,s], as 0/1 float.
// ---------------------------------------------------------------------------
__global__ __launch_bounds__(256) void mask_kernel(
    const unsigned char* __restrict__ amask, const int* __restrict__ cur_index_p,
    float* __restrict__ outm) {
  const int idx = (int)blockIdx.x * 256 + threadIdx.x;
  const int total = B_ * Q_ * S_;
  if (idx >= total) return;
  const int cur = cur_index_p[0];
  const int s = idx & (S_ - 1);
  const bool keep = (s < cur + Q_) && (amask[idx] != 0);
  outm[idx] = keep ? 1.0f : 0.0f;
}

// ---------------------------------------------------------------------------
extern "C" void kernel_launch(void* const* d_in, const int* in_sizes, int n_in,
                              void* d_out, int out_size, void* d_ws,
                              size_t ws_size, hipStream_t stream) {
  (void)in_sizes; (void)n_in; (void)out_size; (void)d_ws; (void)ws_size;

  const float* cached_key   = (const float*)d_in[0];
  const float* cached_value = (const float*)d_in[1];
  const float* key          = (const float*)d_in[2];
  const float* value        = (const float*)d_in[3];
  const unsigned char* am   = (const unsigned char*)d_in[4];
  const int* cur_index      = (const int*)d_in[5];

  float* out = (float*)d_out;
  float* outm = out + (size_t)TOTAL_ROWS * D_;  // after key_deq | value_deq

  kv_requant_kernel<<<NBLOCKS, 256, 0, stream>>>(cached_key, cached_value, key,
                                                 value, cur_index, out);

  const int maskElems = B_ * Q_ * S_;
  mask_kernel<<<(maskElems + 255) / 256, 256, 0, stream>>>(am, cur_index, outm);
}